// SSIM_40346922779448
// MI455X (gfx1250) — compile-verified
//
#include <hip/hip_runtime.h>

typedef float v2f __attribute__((ext_vector_type(2)));
typedef float v8f __attribute__((ext_vector_type(8)));

#define IMG_H 512
#define IMG_W 512
#define OUT_H 502
#define OUT_W 502
#define WSZ   11

#define TILE_OW 64     // output cols per tile
#define TILE_OH 16     // output rows per tile
#define IN_ROWS 32     // 2 x 16 row-chunks (need TILE_OH + 10 = 26)
#define IN_COLS 76     // TILE_OW + 10, +pad for K=28 WMMA reach (48+27+1 = 76)
#define MID_STR 68     // intermediate row stride (64 + pad)
#define TILES_X 8      // ceil(502/64)
#define TILES_Y 32     // ceil(502/16)

// 11-tap Gaussian, sigma=1.5, normalized (matches reference _make_window)
__device__ __constant__ float GK[WSZ] = {
    1.0283801e-3f, 7.5985700e-3f, 3.6000773e-2f, 1.0936043e-1f, 2.1300553e-1f,
    2.6601172e-1f, 2.1300553e-1f, 1.0936043e-1f, 3.6000773e-2f, 7.5985700e-3f,
    1.0283801e-3f};

// Zero-padded band table: GKPAD[15 + t] = GK[t]; index range used: [0, 43].
// bw value for (K-row c, N-col x) is GKPAD[c - x + 15]  (zero outside band).
__device__ __constant__ float GKPAD[44] = {
    0.0f, 0.0f, 0.0f, 0.0f, 0.0f, 0.0f, 0.0f, 0.0f,
    0.0f, 0.0f, 0.0f, 0.0f, 0.0f, 0.0f, 0.0f,
    1.0283801e-3f, 7.5985700e-3f, 3.6000773e-2f, 1.0936043e-1f, 2.1300553e-1f,
    2.6601172e-1f, 2.1300553e-1f, 1.0936043e-1f, 3.6000773e-2f, 7.5985700e-3f,
    1.0283801e-3f,
    0.0f, 0.0f, 0.0f, 0.0f, 0.0f, 0.0f, 0.0f, 0.0f, 0.0f,
    0.0f, 0.0f, 0.0f, 0.0f, 0.0f, 0.0f, 0.0f, 0.0f, 0.0f};

__device__ __forceinline__ unsigned f2key(float f) {
    unsigned b = __float_as_uint(f);
    return (b & 0x80000000u) ? ~b : (b | 0x80000000u);
}
__device__ __forceinline__ float key2f(unsigned k) {
    unsigned b = (k & 0x80000000u) ? (k ^ 0x80000000u) : ~k;
    return __uint_as_float(b);
}

__global__ void k_init(unsigned* wsu) {
    wsu[0] = 0xFFFFFFFFu;  // running min key
    wsu[1] = 0x00000000u;  // running max key
}

__global__ __launch_bounds__(256) void k_minmax(const float* __restrict__ img,
                                                int n, unsigned* __restrict__ wsu) {
    unsigned kmin = 0xFFFFFFFFu, kmax = 0u;
    for (int i = blockIdx.x * blockDim.x + threadIdx.x; i < n;
         i += gridDim.x * blockDim.x) {
        unsigned k = f2key(img[i]);
        kmin = kmin < k ? kmin : k;
        kmax = kmax > k ? kmax : k;
    }
    __shared__ unsigned smin[256], smax[256];
    smin[threadIdx.x] = kmin;
    smax[threadIdx.x] = kmax;
    __syncthreads();
    for (int s = 128; s > 0; s >>= 1) {
        if (threadIdx.x < (unsigned)s) {
            unsigned a = smin[threadIdx.x + s];
            if (a < smin[threadIdx.x]) smin[threadIdx.x] = a;
            unsigned b = smax[threadIdx.x + s];
            if (b > smax[threadIdx.x]) smax[threadIdx.x] = b;
        }
        __syncthreads();
    }
    if (threadIdx.x == 0) {
        atomicMin(&wsu[0], smin[0]);
        atomicMax(&wsu[1], smax[0]);
    }
}

__global__ __launch_bounds__(256) void k_ssim(const float* __restrict__ img1,
                                              const float* __restrict__ img2,
                                              const unsigned* __restrict__ wsu,
                                              float* __restrict__ partial) {
    __shared__ float sIn1[IN_ROWS][IN_COLS];
    __shared__ float sIn2[IN_ROWS][IN_COLS];
    __shared__ float sMid[5][IN_ROWS][MID_STR];
    __shared__ float sRed[256];

    const int row0g = blockIdx.y * TILE_OH;
    const int col0g = blockIdx.x * TILE_OW;
    const float* p1 = img1 + (size_t)blockIdx.z * IMG_H * IMG_W;
    const float* p2 = img2 + (size_t)blockIdx.z * IMG_H * IMG_W;

    // ---- stage input tile into LDS (zero-padded at image edges) ----
    for (int idx = threadIdx.x; idx < IN_ROWS * IN_COLS; idx += 256) {
        int r = idx / IN_COLS, c = idx - r * IN_COLS;
        int gy = row0g + r, gx = col0g + c;
        bool ok = (gy < IMG_H) && (gx < IMG_W);
        size_t off = (size_t)gy * IMG_W + gx;
        sIn1[r][c] = ok ? p1[off] : 0.0f;
        sIn2[r][c] = ok ? p2[off] : 0.0f;
    }
    __syncthreads();

    const int lane = threadIdx.x & 31;
    const int half = lane >> 4;   // 0: lanes 0-15, 1: lanes 16-31
    const int n16  = lane & 15;

    // ---- banded Gaussian B chunks via padded table (loop-invariant loads) ----
    // B layout (4x16 f32): v0 = rows K=2*half, v1 = rows K=2*half+1
    v2f bw[7];
#pragma unroll
    for (int j = 0; j < 7; ++j) {
        const int base = 4 * j + 2 * half - n16 + 15;
        bw[j].x = GKPAD[base];
        bw[j].y = GKPAD[base + 1];
    }

    // ---- horizontal pass on the matrix pipe: one 16x16 chunk per wave ----
    {
        const int wave = threadIdx.x >> 5;        // 0..7
        const int row0 = (wave >> 2) * 16;        // 0 or 16
        const int col0 = (wave & 3) * 16;         // 0,16,32,48
        const int my = row0 + n16;                // A-matrix row for this lane
        v8f acc0 = {}, acc1 = {}, acc2 = {}, acc3 = {}, acc4 = {};
#pragma unroll
        for (int j = 0; j < 7; ++j) {
            const int cb = col0 + 4 * j + 2 * half;   // A: K=2*half,2*half+1
            v2f a1, a2;
            a1.x = sIn1[my][cb];  a1.y = sIn1[my][cb + 1];
            a2.x = sIn2[my][cb];  a2.y = sIn2[my][cb + 1];
            v2f a11 = a1 * a1;
            v2f a22 = a2 * a2;
            v2f a12 = a1 * a2;
            acc0 = __builtin_amdgcn_wmma_f32_16x16x4_f32(false, a1,  false, bw[j], (short)0, acc0, false, false);
            acc1 = __builtin_amdgcn_wmma_f32_16x16x4_f32(false, a2,  false, bw[j], (short)0, acc1, false, false);
            acc2 = __builtin_amdgcn_wmma_f32_16x16x4_f32(false, a11, false, bw[j], (short)0, acc2, false, false);
            acc3 = __builtin_amdgcn_wmma_f32_16x16x4_f32(false, a22, false, bw[j], (short)0, acc3, false, false);
            acc4 = __builtin_amdgcn_wmma_f32_16x16x4_f32(false, a12, false, bw[j], (short)0, acc4, false, false);
        }
        // D layout: VGPR v holds M = v + 8*half, N = n16
#pragma unroll
        for (int v = 0; v < 8; ++v) {
            const int rr = row0 + v + 8 * half;
            const int cc = col0 + n16;
            sMid[0][rr][cc] = acc0[v];
            sMid[1][rr][cc] = acc1[v];
            sMid[2][rr][cc] = acc2[v];
            sMid[3][rr][cc] = acc3[v];
            sMid[4][rr][cc] = acc4[v];
        }
    }
    __syncthreads();

    // ---- dynamic range constants (resolved by k_minmax) ----
    const float fmin = key2f(wsu[0]);
    const float fmax = key2f(wsu[1]);
    const float max_val = fmax > 128.0f ? 255.0f : 1.0f;
    const float min_val = fmin < -0.5f ? -1.0f : 0.0f;
    const float Lr = max_val - min_val;
    const float C1 = (0.01f * Lr) * (0.01f * Lr);
    const float C2 = (0.03f * Lr) * (0.03f * Lr);

    // ---- vertical 11-tap + SSIM map (VALU), 4 outputs per thread ----
    const int tx  = threadIdx.x & 63;
    const int ty0 = (threadIdx.x >> 6) * 4;
    float res[5][4];
#pragma unroll
    for (int m = 0; m < 5; ++m) {
        float win[14];
#pragma unroll
        for (int i = 0; i < 14; ++i) win[i] = sMid[m][ty0 + i][tx];
#pragma unroll
        for (int k = 0; k < 4; ++k) {
            float s = 0.0f;
#pragma unroll
            for (int t = 0; t < WSZ; ++t) s = fmaf(win[k + t], GK[t], s);
            res[m][k] = s;
        }
    }
    float lsum = 0.0f;
#pragma unroll
    for (int k = 0; k < 4; ++k) {
        const int gy = row0g + ty0 + k;
        const int gx = col0g + tx;
        if (gy < OUT_H && gx < OUT_W) {
            const float mu1 = res[0][k], mu2 = res[1][k];
            const float mu1s = mu1 * mu1, mu2s = mu2 * mu2, mu12 = mu1 * mu2;
            const float s1  = res[2][k] - mu1s;
            const float s2  = res[3][k] - mu2s;
            const float s12 = res[4][k] - mu12;
            const float v1 = 2.0f * s12 + C2;
            const float v2 = s1 + s2 + C2;
            const float num = (2.0f * mu12 + C1) * v1;
            const float den = (mu1s + mu2s + C1) * v2;
            lsum += num / den;
        }
    }

    // ---- deterministic block reduction ----
    sRed[threadIdx.x] = lsum;
    __syncthreads();
    for (int s = 128; s > 0; s >>= 1) {
        if (threadIdx.x < (unsigned)s) sRed[threadIdx.x] += sRed[threadIdx.x + s];
        __syncthreads();
    }
    if (threadIdx.x == 0) {
        const int bid = (blockIdx.z * gridDim.y + blockIdx.y) * gridDim.x + blockIdx.x;
        partial[bid] = sRed[0];
    }
}

__global__ __launch_bounds__(256) void k_final(const float* __restrict__ partial,
                                               int nblk, float inv_cnt,
                                               float* __restrict__ out) {
    __shared__ float s[256];
    float a = 0.0f;
    for (int i = threadIdx.x; i < nblk; i += 256) a += partial[i];  // fixed order
    s[threadIdx.x] = a;
    __syncthreads();
    for (int st = 128; st > 0; st >>= 1) {
        if (threadIdx.x < (unsigned)st) s[threadIdx.x] += s[threadIdx.x + st];
        __syncthreads();
    }
    if (threadIdx.x == 0) out[0] = s[0] * inv_cnt;
}

extern "C" void kernel_launch(void* const* d_in, const int* in_sizes, int n_in,
                              void* d_out, int out_size, void* d_ws, size_t ws_size,
                              hipStream_t stream) {
    const float* img1 = (const float*)d_in[0];
    const float* img2 = (const float*)d_in[1];
    float* out = (float*)d_out;

    unsigned* wsu = (unsigned*)d_ws;
    float* partial = (float*)d_ws + 2;

    const int n = in_sizes[0];                       // 32*3*512*512
    const int planes = n / (IMG_H * IMG_W);          // 96
    const int nblk = TILES_X * TILES_Y * planes;     // 24576
    const double cnt = (double)planes * (double)OUT_H * (double)OUT_W;
    const float inv_cnt = (float)(1.0 / cnt);

    k_init<<<1, 1, 0, stream>>>(wsu);
    k_minmax<<<1024, 256, 0, stream>>>(img1, n, wsu);
    dim3 grid(TILES_X, TILES_Y, planes);
    k_ssim<<<grid, 256, 0, stream>>>(img1, img2, wsu, partial);
    k_final<<<1, 256, 0, stream>>>(partial, nblk, inv_cnt, out);
}